// LESSFormer_61538291417293
// MI455X (gfx1250) — compile-verified
//
#include <hip/hip_runtime.h>
#include <hip/hip_bf16.h>
#include <math.h>

#define B_     2
#define N_     4096
#define DIM_   64
#define INNER_ 256
#define DEPTH_ 2
#define EPS_   1e-5f

typedef __attribute__((ext_vector_type(2))) float v2f;
typedef __attribute__((ext_vector_type(8))) float v8f;
typedef unsigned long long u64;

// ---------------- LayerNorm: one wave32 per 64-element row ----------------
__global__ void ln_kernel(const float* __restrict__ in, const float* __restrict__ g,
                          const float* __restrict__ b, float* __restrict__ out, int rows) {
  int wid  = (int)((blockIdx.x * blockDim.x + threadIdx.x) >> 5);
  int lane = threadIdx.x & 31;
  if (wid >= rows) return;
  const float* r = in + (size_t)wid * DIM_;
  float x0 = r[lane * 2], x1 = r[lane * 2 + 1];
  float s = x0 + x1, sq = x0 * x0 + x1 * x1;
  #pragma unroll
  for (int off = 16; off > 0; off >>= 1) {
    s  += __shfl_xor(s,  off, 32);
    sq += __shfl_xor(sq, off, 32);
  }
  float mean = s * (1.0f / DIM_);
  float var  = sq * (1.0f / DIM_) - mean * mean;
  float inv  = rsqrtf(var + EPS_);
  float* o = out + (size_t)wid * DIM_;
  o[lane * 2]     = (x0 - mean) * inv * g[lane * 2]     + b[lane * 2];
  o[lane * 2 + 1] = (x1 - mean) * inv * g[lane * 2 + 1] + b[lane * 2 + 1];
}

// ------------- Generic register-blocked fp32 WMMA GEMM (small projections) -------------
__global__ __launch_bounds__(256)
void wmma_gemm_kernel(const float* __restrict__ A, int lda,
                      const float* __restrict__ Bm, int ldb, int transB,
                      float* __restrict__ C, int ldc,
                      int M, int Nc, int K, float alpha,
                      const float* __restrict__ bias,
                      const float* __restrict__ resid, int ldr,
                      int gelu) {
  const int ntx   = Nc >> 6;
  const int tiles = (M >> 5) * ntx;
  int t = blockIdx.x * (blockDim.x >> 5) + (threadIdx.x >> 5);
  if (t >= tiles) return;                     // wave-uniform (EXEC all-1s for WMMA)
  int tm = t / ntx, tn = t - tm * ntx;
  int lane = threadIdx.x & 31;
  int l16  = lane & 15;
  int hi   = lane >> 4;
  int m0   = tm << 5;
  int n0   = tn << 6;

  const float* a0 = A + (size_t)(m0 + l16) * lda + 2 * hi;
  const float* a1 = a0 + (size_t)16 * lda;

  v8f acc[2][4];
  #pragma unroll
  for (int i = 0; i < 2; ++i)
    #pragma unroll
    for (int j = 0; j < 4; ++j) acc[i][j] = {};

  if (transB) {
    const float* b0 = Bm + (size_t)(n0 + l16) * ldb + 2 * hi;
    #pragma unroll 2
    for (int k0 = 0; k0 < K; k0 += 4) {
      v2f a[2], b[4];
      a[0] = *(const v2f*)(a0 + k0);
      a[1] = *(const v2f*)(a1 + k0);
      #pragma unroll
      for (int j = 0; j < 4; ++j)
        b[j] = *(const v2f*)(b0 + (size_t)(j * 16) * ldb + k0);
      #pragma unroll
      for (int i = 0; i < 2; ++i)
        #pragma unroll
        for (int j = 0; j < 4; ++j)
          acc[i][j] = __builtin_amdgcn_wmma_f32_16x16x4_f32(false, a[i], false, b[j],
                                                            (short)0, acc[i][j], false, false);
    }
  } else {
    const float* bbase = Bm + (size_t)(2 * hi) * ldb + n0 + l16;
    #pragma unroll 2
    for (int k0 = 0; k0 < K; k0 += 4) {
      v2f a[2], b[4];
      a[0] = *(const v2f*)(a0 + k0);
      a[1] = *(const v2f*)(a1 + k0);
      const float* bk = bbase + (size_t)k0 * ldb;
      #pragma unroll
      for (int j = 0; j < 4; ++j) {
        b[j].x = bk[j * 16];
        b[j].y = bk[j * 16 + ldb];
      }
      #pragma unroll
      for (int i = 0; i < 2; ++i)
        #pragma unroll
        for (int j = 0; j < 4; ++j)
          acc[i][j] = __builtin_amdgcn_wmma_f32_16x16x4_f32(false, a[i], false, b[j],
                                                            (short)0, acc[i][j], false, false);
    }
  }

  int colb = n0 + l16;
  float bv[4];
  #pragma unroll
  for (int j = 0; j < 4; ++j) bv[j] = bias ? bias[colb + j * 16] : 0.0f;

  #pragma unroll
  for (int i = 0; i < 2; ++i)
    #pragma unroll
    for (int j = 0; j < 4; ++j)
      #pragma unroll
      for (int r = 0; r < 8; ++r) {
        int row = m0 + i * 16 + r + 8 * hi;
        int col = colb + j * 16;
        float v = acc[i][j][r] * alpha + bv[j];
        if (gelu) v = 0.5f * v * (1.0f + erff(v * 0.70710678118f));
        if (resid) v += resid[(size_t)row * ldr + col];
        C[(size_t)row * ldc + col] = v;
      }
}

// ------------- S = alpha * Q @ K^T, async-LDS double-buffered pipeline -------------
#define KBLK_  16
#define LPAD_  20                              // floats per LDS row (16 + 4 pad)

__global__ __launch_bounds__(256)
void qk_gemm_kernel(const float* __restrict__ Q, const float* __restrict__ Km,
                    float* __restrict__ S, float alpha) {
  __shared__ float At[2][128 * LPAD_];
  __shared__ float Bt[2][128 * LPAD_];

  const int tid  = threadIdx.x;
  const int m0b  = blockIdx.x * 128;
  const int n0b  = blockIdx.y * 128;
  const int wave = tid >> 5;
  const int wm   = wave & 3;
  const int wn   = wave >> 2;
  const int lane = tid & 31;
  const int l16  = lane & 15;
  const int hi   = lane >> 4;

  const int c0 = tid, c1 = tid + 256;
  const int r0 = c0 >> 2, f0 = (c0 & 3) * 4;
  const int r1 = c1 >> 2, f1 = (c1 & 3) * 4;

  const u64 qa0 = (u64)(uintptr_t)(Q + (size_t)(m0b + r0) * 768 + f0);
  const u64 qa1 = (u64)(uintptr_t)(Q + (size_t)(m0b + r1) * 768 + f1);
  const u64 ka0 = (u64)(uintptr_t)(Km + (size_t)(n0b + r0) * 768 + f0);
  const u64 ka1 = (u64)(uintptr_t)(Km + (size_t)(n0b + r1) * 768 + f1);
  const unsigned la0 = (unsigned)(uintptr_t)&At[0][0] + (unsigned)(r0 * LPAD_ + f0) * 4u;
  const unsigned la1 = (unsigned)(uintptr_t)&At[0][0] + (unsigned)(r1 * LPAD_ + f1) * 4u;
  const unsigned lb0 = (unsigned)(uintptr_t)&Bt[0][0] + (unsigned)(r0 * LPAD_ + f0) * 4u;
  const unsigned lb1 = (unsigned)(uintptr_t)&Bt[0][0] + (unsigned)(r1 * LPAD_ + f1) * 4u;
  const unsigned bufStride = (unsigned)(128 * LPAD_ * 4);

  auto stage = [&](int kb, int buf) {
    u64 ko = (u64)kb * KBLK_ * 4u;
    unsigned bo = (unsigned)buf * bufStride;
    asm volatile("global_load_async_to_lds_b128 %0, %1, off"
                 :: "v"(la0 + bo), "v"(qa0 + ko) : "memory");
    asm volatile("global_load_async_to_lds_b128 %0, %1, off"
                 :: "v"(la1 + bo), "v"(qa1 + ko) : "memory");
    asm volatile("global_load_async_to_lds_b128 %0, %1, off"
                 :: "v"(lb0 + bo), "v"(ka0 + ko) : "memory");
    asm volatile("global_load_async_to_lds_b128 %0, %1, off"
                 :: "v"(lb1 + bo), "v"(ka1 + ko) : "memory");
  };

  v8f acc[2][4];
  #pragma unroll
  for (int i = 0; i < 2; ++i)
    #pragma unroll
    for (int j = 0; j < 4; ++j) acc[i][j] = {};

  stage(0, 0);

  const int nStages = INNER_ / KBLK_;          // 16
  for (int kb = 0; kb < nStages; ++kb) {
    int buf = kb & 1;
    asm volatile("s_wait_asynccnt 0x0" ::: "memory");
    __syncthreads();
    if (kb + 1 < nStages) stage(kb + 1, buf ^ 1);

    const float* Ab = &At[buf][0];
    const float* Bb = &Bt[buf][0];
    #pragma unroll
    for (int kk = 0; kk < KBLK_; kk += 4) {
      v2f a[2], b[4];
      a[0] = *(const v2f*)(Ab + (wm * 32 + l16) * LPAD_ + kk + 2 * hi);
      a[1] = *(const v2f*)(Ab + (wm * 32 + 16 + l16) * LPAD_ + kk + 2 * hi);
      #pragma unroll
      for (int j = 0; j < 4; ++j)
        b[j] = *(const v2f*)(Bb + (wn * 64 + j * 16 + l16) * LPAD_ + kk + 2 * hi);
      #pragma unroll
      for (int i = 0; i < 2; ++i)
        #pragma unroll
        for (int j = 0; j < 4; ++j)
          acc[i][j] = __builtin_amdgcn_wmma_f32_16x16x4_f32(false, a[i], false, b[j],
                                                            (short)0, acc[i][j], false, false);
    }
  }

  #pragma unroll
  for (int i = 0; i < 2; ++i)
    #pragma unroll
    for (int j = 0; j < 4; ++j)
      #pragma unroll
      for (int r = 0; r < 8; ++r) {
        int row = m0b + wm * 32 + i * 16 + r + 8 * hi;
        int col = n0b + wn * 64 + j * 16 + l16;
        S[(size_t)row * N_ + col] = acc[i][j][r] * alpha;
      }
}

// ------------- P[ks] = attn[:, ks-chunk] @ V[ks-chunk, :], async-LDS + K-split -------------
// Block tile 128x128, 8 waves (4M x 2N) of 32x64; KBLK=16 double-buffered stages.
// attn panel row-major (pad 20); V panel K-major, rows padded to 136 floats so the two
// half-wave K-rows hit disjoint LDS bank ranges.
#define AVKS_   4                              // deterministic K splits
#define AVKC_   (N_ / AVKS_)                   // 1024 K per split
#define BPADAV_ 136

__global__ __launch_bounds__(256)
void av_gemm_kernel(const float* __restrict__ Amat, const float* __restrict__ V,
                    float* __restrict__ P) {
  __shared__ float At[2][128 * LPAD_];
  __shared__ float Bt[2][KBLK_ * BPADAV_];

  const int tid  = threadIdx.x;
  const int m0b  = blockIdx.x * 128;
  const int n0b  = blockIdx.y * 128;
  const int k0b  = blockIdx.z * AVKC_;
  const int wave = tid >> 5;
  const int wm   = wave & 3;
  const int wn   = wave >> 2;
  const int lane = tid & 31;
  const int l16  = lane & 15;
  const int hi   = lane >> 4;

  const int c0 = tid, c1 = tid + 256;
  const int ar0 = c0 >> 2, af0 = (c0 & 3) * 4;     // A: 128 rows x 4 chunks
  const int ar1 = c1 >> 2, af1 = (c1 & 3) * 4;
  const int br0 = c0 >> 5, bf0 = (c0 & 31) * 4;    // B: 16 rows x 32 chunks
  const int br1 = c1 >> 5, bf1 = (c1 & 31) * 4;

  const u64 ga0 = (u64)(uintptr_t)(Amat + (size_t)(m0b + ar0) * N_ + k0b + af0);
  const u64 ga1 = (u64)(uintptr_t)(Amat + (size_t)(m0b + ar1) * N_ + k0b + af1);
  const u64 gb0 = (u64)(uintptr_t)(V + (size_t)(k0b + br0) * 768 + n0b + bf0);
  const u64 gb1 = (u64)(uintptr_t)(V + (size_t)(k0b + br1) * 768 + n0b + bf1);
  const unsigned laA0 = (unsigned)(uintptr_t)&At[0][0] + (unsigned)(ar0 * LPAD_ + af0) * 4u;
  const unsigned laA1 = (unsigned)(uintptr_t)&At[0][0] + (unsigned)(ar1 * LPAD_ + af1) * 4u;
  const unsigned laB0 = (unsigned)(uintptr_t)&Bt[0][0] + (unsigned)(br0 * BPADAV_ + bf0) * 4u;
  const unsigned laB1 = (unsigned)(uintptr_t)&Bt[0][0] + (unsigned)(br1 * BPADAV_ + bf1) * 4u;
  const unsigned bufA = (unsigned)(128 * LPAD_ * 4);
  const unsigned bufB = (unsigned)(KBLK_ * BPADAV_ * 4);

  auto stage = [&](int kb, int buf) {
    u64 koA = (u64)kb * KBLK_ * 4u;                 // advance along attn row
    u64 koB = (u64)kb * KBLK_ * 768u * 4u;          // advance V rows
    unsigned boA = (unsigned)buf * bufA;
    unsigned boB = (unsigned)buf * bufB;
    asm volatile("global_load_async_to_lds_b128 %0, %1, off"
                 :: "v"(laA0 + boA), "v"(ga0 + koA) : "memory");
    asm volatile("global_load_async_to_lds_b128 %0, %1, off"
                 :: "v"(laA1 + boA), "v"(ga1 + koA) : "memory");
    asm volatile("global_load_async_to_lds_b128 %0, %1, off"
                 :: "v"(laB0 + boB), "v"(gb0 + koB) : "memory");
    asm volatile("global_load_async_to_lds_b128 %0, %1, off"
                 :: "v"(laB1 + boB), "v"(gb1 + koB) : "memory");
  };

  v8f acc[2][4];
  #pragma unroll
  for (int i = 0; i < 2; ++i)
    #pragma unroll
    for (int j = 0; j < 4; ++j) acc[i][j] = {};

  stage(0, 0);

  const int nStages = AVKC_ / KBLK_;               // 64
  for (int kb = 0; kb < nStages; ++kb) {
    int buf = kb & 1;
    asm volatile("s_wait_asynccnt 0x0" ::: "memory");
    __syncthreads();
    if (kb + 1 < nStages) stage(kb + 1, buf ^ 1);

    const float* Ab = &At[buf][0];
    const float* Bb = &Bt[buf][0];
    #pragma unroll
    for (int kk = 0; kk < KBLK_; kk += 4) {
      v2f a[2], b[4];
      a[0] = *(const v2f*)(Ab + (wm * 32 + l16) * LPAD_ + kk + 2 * hi);
      a[1] = *(const v2f*)(Ab + (wm * 32 + 16 + l16) * LPAD_ + kk + 2 * hi);
      const float* brow = Bb + (kk + 2 * hi) * BPADAV_ + wn * 64 + l16;
      #pragma unroll
      for (int j = 0; j < 4; ++j) {
        b[j].x = brow[j * 16];
        b[j].y = brow[j * 16 + BPADAV_];
      }
      #pragma unroll
      for (int i = 0; i < 2; ++i)
        #pragma unroll
        for (int j = 0; j < 4; ++j)
          acc[i][j] = __builtin_amdgcn_wmma_f32_16x16x4_f32(false, a[i], false, b[j],
                                                            (short)0, acc[i][j], false, false);
    }
  }

  float* Pz = P + (size_t)blockIdx.z * N_ * INNER_;
  #pragma unroll
  for (int i = 0; i < 2; ++i)
    #pragma unroll
    for (int j = 0; j < 4; ++j)
      #pragma unroll
      for (int r = 0; r < 8; ++r) {
        int row = m0b + wm * 32 + i * 16 + r + 8 * hi;
        int col = n0b + wn * 64 + j * 16 + l16;
        Pz[(size_t)row * INNER_ + col] = acc[i][j][r];
      }
}

// Deterministic fixed-order sum of the 4 K-split partials
__global__ void ksum_kernel(const float* __restrict__ P, float* __restrict__ out, int n) {
  int i = blockIdx.x * blockDim.x + threadIdx.x;
  if (i >= n) return;
  out[i] = ((P[i] + P[i + (size_t)n]) + P[i + 2 * (size_t)n]) + P[i + 3 * (size_t)n];
}

// ------------- Row softmax over N_=4096, optionally adjacency-masked -------------
__global__ void softmax_kernel(const float* __restrict__ S, const float* __restrict__ Adj,
                               float* __restrict__ out, int masked) {
  __shared__ float shm[8], shs[8];
  int row = blockIdx.x;
  int tid = threadIdx.x;
  const float* srow = S + (size_t)row * N_;
  float vals[16];
  float lmax = -3.0e38f;
  #pragma unroll
  for (int i = 0; i < 16; ++i) {
    int col = i * 256 + tid;
    float v = srow[col];
    if (masked) {
      float a = Adj[(size_t)row * N_ + col];
      v = (a > 0.0f) ? v : -9.0e15f;
      if (col == row) v = 1.0f;
    }
    vals[i] = v;
    lmax = fmaxf(lmax, v);
  }
  #pragma unroll
  for (int off = 16; off > 0; off >>= 1) lmax = fmaxf(lmax, __shfl_xor(lmax, off, 32));
  if ((tid & 31) == 0) shm[tid >> 5] = lmax;
  __syncthreads();
  float rmax = shm[0];
  #pragma unroll
  for (int i = 1; i < 8; ++i) rmax = fmaxf(rmax, shm[i]);

  float lsum = 0.0f;
  #pragma unroll
  for (int i = 0; i < 16; ++i) { vals[i] = __expf(vals[i] - rmax); lsum += vals[i]; }
  #pragma unroll
  for (int off = 16; off > 0; off >>= 1) lsum += __shfl_xor(lsum, off, 32);
  if ((tid & 31) == 0) shs[tid >> 5] = lsum;
  __syncthreads();
  float rsum = 0.0f;
  #pragma unroll
  for (int i = 0; i < 8; ++i) rsum += shs[i];
  float inv = 1.0f / rsum;

  float* orow = out + (size_t)row * N_;
  #pragma unroll
  for (int i = 0; i < 16; ++i) orow[i * 256 + tid] = vals[i] * inv;
}

// ---------------------------------------------------------------------------
static inline void gemm(hipStream_t s, const float* A, int lda, const float* B, int ldb,
                        int transB, float* C, int ldc, int M, int N, int K, float alpha,
                        const float* bias, const float* resid, int ldr, int gelu_flag) {
  int tiles  = (M / 32) * (N / 64);
  int blocks = (tiles + 7) / 8;
  wmma_gemm_kernel<<<blocks, 256, 0, s>>>(A, lda, B, ldb, transB, C, ldc,
                                          M, N, K, alpha, bias, resid, ldr, gelu_flag);
}

extern "C" void kernel_launch(void* const* d_in, const int* in_sizes, int n_in,
                              void* d_out, int out_size, void* d_ws, size_t ws_size,
                              hipStream_t stream) {
  const float* embed = (const float*)d_in[0];
  const float* Adj   = (const float*)d_in[1];
  const float* g1    = (const float*)d_in[2];
  const float* b1    = (const float*)d_in[3];
  const float* Wqkv  = (const float*)d_in[4];
  const float* Wout  = (const float*)d_in[5];
  const float* bout  = (const float*)d_in[6];
  const float* g2    = (const float*)d_in[7];
  const float* b2    = (const float*)d_in[8];
  const float* W1    = (const float*)d_in[9];
  const float* bb1   = (const float*)d_in[10];
  const float* W2    = (const float*)d_in[11];
  const float* bb2   = (const float*)d_in[12];
  const float* gf    = (const float*)d_in[13];
  const float* bf    = (const float*)d_in[14];

  // Output tuple layout: x [B,N,DIM] | attn [1,1,N,N] | dots_softmax [N,N]
  float* out_x    = (float*)d_out;
  float* out_attn = out_x + (size_t)B_ * N_ * DIM_;
  float* out_dots = out_attn + (size_t)N_ * N_;

  // Workspace (~59 MB fp32)
  float* w    = (float*)d_ws;
  float* xn   = w;  w += (size_t)B_ * N_ * DIM_;
  float* x    = w;  w += (size_t)B_ * N_ * DIM_;
  float* xm   = w;  w += (size_t)B_ * N_ * DIM_;
  float* h    = w;  w += (size_t)B_ * N_ * DIM_;
  float* qkv  = w;  w += (size_t)B_ * N_ * (3 * INNER_);
  float* ob   = w;  w += (size_t)B_ * N_ * INNER_;
  float* part = w;  w += (size_t)AVKS_ * N_ * INNER_;

  const int rows    = B_ * N_;
  const int lnBlks  = rows / 8;
  const float scale = 0.125f;                  // DIM^-0.5

  const float* xcur = embed;
  for (int l = 0; l < DEPTH_; ++l) {
    // xn = LN(x; g1,b1)
    ln_kernel<<<lnBlks, 256, 0, stream>>>(xcur, g1 + l * DIM_, b1 + l * DIM_, xn, rows);

    // qkv = xn @ Wqkv[l]   [8192,64]x[64,768]
    gemm(stream, xn, DIM_, Wqkv + (size_t)l * DIM_ * 3 * INNER_, 3 * INNER_, 0,
         qkv, 3 * INNER_, rows, 3 * INNER_, DIM_, 1.0f, nullptr, nullptr, 0, 0);

    // S = scale * q0 @ k0^T (batch 0 only) -> d_out dots region; async-LDS pipeline
    {
      dim3 grid(N_ / 128, N_ / 128);
      qk_gemm_kernel<<<grid, 256, 0, stream>>>(qkv, qkv + INNER_, out_dots, scale);
    }

    // attn = softmax(mask(S)) -> d_out attn region
    softmax_kernel<<<N_, 256, 0, stream>>>(out_dots, Adj, out_attn, 1);

    // ob[b] = attn @ v[b] via async-LDS K-split partials + deterministic reduction
    for (int b = 0; b < B_; ++b) {
      dim3 g(N_ / 128, INNER_ / 128, AVKS_);
      av_gemm_kernel<<<g, 256, 0, stream>>>(out_attn,
                                            qkv + (size_t)b * N_ * 3 * INNER_ + 2 * INNER_,
                                            part);
      ksum_kernel<<<(N_ * INNER_) / 256, 256, 0, stream>>>(part, ob + (size_t)b * N_ * INNER_,
                                                           N_ * INNER_);
    }

    // x = xn + ob @ Wout[l] + bout[l]
    gemm(stream, ob, INNER_, Wout + (size_t)l * INNER_ * DIM_, DIM_, 0,
         x, DIM_, rows, DIM_, INNER_, 1.0f, bout + l * DIM_, xn, DIM_, 0);

    // xm = LN(x; g2,b2)
    ln_kernel<<<lnBlks, 256, 0, stream>>>(x, g2 + l * DIM_, b2 + l * DIM_, xm, rows);

    // h = gelu(xm @ W1[l] + bb1[l])
    gemm(stream, xm, DIM_, W1 + (size_t)l * DIM_ * DIM_, DIM_, 0,
         h, DIM_, rows, DIM_, DIM_, 1.0f, bb1 + l * DIM_, nullptr, 0, 1);

    // x = xm + h @ W2[l] + bb2[l]
    gemm(stream, h, DIM_, W2 + (size_t)l * DIM_ * DIM_, DIM_, 0,
         x, DIM_, rows, DIM_, DIM_, 1.0f, bb2 + l * DIM_, xm, DIM_, 0);

    xcur = x;
  }

  // dots_softmax = softmax(S_last), in place (attn kernel already consumed S)
  softmax_kernel<<<N_, 256, 0, stream>>>(out_dots, nullptr, out_dots, 0);

  // final: out_x = LN(x; gf, bf)
  ln_kernel<<<lnBlks, 256, 0, stream>>>(x, gf, bf, out_x, rows);
}